// HashEncodingEnsembleParallel_33036888441133
// MI455X (gfx1250) — compile-verified
//
#include <hip/hip_runtime.h>

typedef float v2f __attribute__((ext_vector_type(2)));
typedef float v8f __attribute__((ext_vector_type(8)));

#define NLEV   14
#define LOG2T  19
#define TSIZE  (1u << LOG2T)
#define ENS    8
#define NFEATC 2
#define FDIM   16

// ds_swizzle group-of-32 xor pattern: offset[14:10]=xor_mask, [9:5]=or=0, [4:0]=and=0x1f
__device__ __forceinline__ float swz_xor(float v, int mask_imm_times_1024_plus_1f_unused) {
  return v; // placeholder (not used)
}

__device__ __forceinline__ float swizzle_xor1(float v) {
  return __int_as_float(__builtin_amdgcn_ds_swizzle(__float_as_int(v), (1 << 10) | 0x1f));
}
__device__ __forceinline__ float swizzle_xor2(float v) {
  return __int_as_float(__builtin_amdgcn_ds_swizzle(__float_as_int(v), (2 << 10) | 0x1f));
}
__device__ __forceinline__ float swizzle_xor4(float v) {
  return __int_as_float(__builtin_amdgcn_ds_swizzle(__float_as_int(v), (4 << 10) | 0x1f));
}

__global__ __launch_bounds__(256) void hashenc_ens_kernel(
    const float* __restrict__ x,      // [N,3]
    const float* __restrict__ code,   // [N,8]
    const float* __restrict__ table,  // [14, 2^19, 16]
    float* __restrict__ out)          // [N, 28]
{
  const int tid  = threadIdx.x;
  const int lane = tid & 31;
  const int wave = tid >> 5;
  const int c    = lane & 7;   // corner (for A) / ensemble index (after WMMA)
  const int grp  = lane >> 3;  // which of the 4 samples this lane's result belongs to
  const int l    = blockIdx.y;
  const int n_base = (blockIdx.x * 8 + wave) * 4;

  // ---- level constants, matching reference float64 pow -> float32 cast -> ceil ----
  double pl = 1.0;
  for (int i = 0; i < l; ++i) pl *= 1.4472692012786865;
  const float    scale = (float)(16.0 * pl - 1.0);
  const unsigned res   = (unsigned)ceilf(scale) + 1u;
  const bool     dense = ((unsigned long long)res * res * res) <= (unsigned long long)TSIZE;

  // Block-diagonal packing: sample q's 8x8 outer product lands in D block owned by
  // lanes 8q..8q+7.  A-column roles: grp0->P0(k=0), grp1->P2(k=1), grp2->P1(k=2),
  // grp3->P3(k=3).  B rows: grp0 supplies code(P0,P2) on k=0/1, grp3 supplies
  // code(P1,P3) on k=2/3.
  const int cwPair  = (grp == 1) ? 2 : (grp == 2) ? 1 : grp;
  const int idxPair = grp;

  // ---- smoothstep weights for the sample whose cw this lane feeds into A ----
  float w[3];
  {
    const float* xs = x + (size_t)(n_base + cwPair) * 3;
#pragma unroll
    for (int d = 0; d < 3; ++d) {
      float pos = xs[d] * scale + 0.5f;
      float g   = floorf(pos);
      float t   = pos - g;
      w[d] = t * t * (3.0f - 2.0f * t);
    }
  }
  float cw = 1.0f;
#pragma unroll
  for (int d = 0; d < 3; ++d)
    cw *= ((c >> d) & 1) ? w[d] : (1.0f - w[d]);

  // A (16x4 f32): lane m<16 holds {A[m,0],A[m,1]}, lane m+16 holds {A[m,2],A[m,3]}
  v2f A;
  A.x = ((grp & 1) == 0) ? cw : 0.0f;
  A.y = ((grp & 1) != 0) ? cw : 0.0f;

  // B (4x16 f32): lane j<16 holds {B[0,j],B[1,j]}, lane j+16 holds {B[2,j],B[3,j]}
  const int nb0 = n_base + ((grp == 3) ? 1 : 0);
  const int nb1 = n_base + ((grp == 3) ? 3 : 2);
  const float cb0 = code[(size_t)nb0 * ENS + c];
  const float cb1 = code[(size_t)nb1 * ENS + c];
  v2f B;
  B.x = (grp == 0 || grp == 3) ? cb0 : 0.0f;
  B.y = (grp == 0 || grp == 3) ? cb1 : 0.0f;

  // p[c,e] = cw[c]*code[e] for 4 samples in one exact-f32 WMMA
  v8f Cz = {};
  v8f D  = __builtin_amdgcn_wmma_f32_16x16x4_f32(false, A, false, B, (short)0, Cz,
                                                 false, false);
  // Lane (grp,c) now holds p_grp[corner r, e=c] in D[r], r=0..7.

  // ---- corner base coordinates for this lane's own sample ----
  unsigned gu[3];
  {
    const float* xs = x + (size_t)(n_base + idxPair) * 3;
#pragma unroll
    for (int d = 0; d < 3; ++d) {
      float pos = xs[d] * scale + 0.5f;
      gu[d] = (unsigned)floorf(pos);
    }
  }

  // ---- gather 8 corners: lane e reads bytes [8e,8e+8) of each 64B row ----
  const float2* tab = (const float2*)(table + (size_t)l * TSIZE * FDIM);
  float ax = 0.0f, ay = 0.0f;
#pragma unroll
  for (int r = 0; r < 8; ++r) {
    unsigned ux = gu[0] + (r & 1);
    unsigned uy = gu[1] + ((r >> 1) & 1);
    unsigned uz = gu[2] + ((r >> 2) & 1);
    unsigned h  = ux ^ (uy * 2654435761u) ^ (uz * 805459861u);
    unsigned di = ux + uy * res + uz * res * res;
    unsigned idx = (dense ? di : h) & (TSIZE - 1u);
    float2 f = tab[(size_t)idx * (FDIM / 2) + c];
    float  pw = D[r];
    ax = fmaf(pw, f.x, ax);
    ay = fmaf(pw, f.y, ay);
  }

  // ---- butterfly reduce over the 8 ensemble lanes of this group (ds_swizzle) ----
  ax += swizzle_xor1(ax);  ay += swizzle_xor1(ay);
  ax += swizzle_xor2(ax);  ay += swizzle_xor2(ay);
  ax += swizzle_xor4(ax);  ay += swizzle_xor4(ay);

  if (c == 0) {
    float2* o = (float2*)(out + (size_t)(n_base + grp) * (NLEV * NFEATC) + l * NFEATC);
    *o = make_float2(ax, ay);
  }
}

extern "C" void kernel_launch(void* const* d_in, const int* in_sizes, int n_in,
                              void* d_out, int out_size, void* d_ws, size_t ws_size,
                              hipStream_t stream) {
  const float* x     = (const float*)d_in[0];
  const float* code  = (const float*)d_in[1];
  const float* table = (const float*)d_in[2];
  float* out = (float*)d_out;
  const int n = in_sizes[0] / 3;          // 131072 samples
  dim3 grid(n / 32, NLEV);                // 8 waves/block * 4 samples/wave = 32
  hipLaunchKernelGGL(hashenc_ens_kernel, grid, dim3(256), 0, stream,
                     x, code, table, out);
}